// CrossPointInteraction_81870666596679
// MI455X (gfx1250) — compile-verified
//
#include <hip/hip_runtime.h>
#include <hip/hip_bf16.h>
#include <math.h>

typedef __attribute__((ext_vector_type(2))) float v2f;
typedef __attribute__((ext_vector_type(8))) float v8f;

// Problem constants (from setup_inputs)
constexpr int B_ = 2, C_ = 3, D_ = 64, N_ = 4096, K_ = 64;
constexpr int F_ = 134;            // 2C + 2D fused input channels
constexpr int S_ = 34;             // K-steps of 4 covering 136 (padded) channels
constexpr float EPS_ = 1e-5f;

// Workspace layout (float offsets)
constexpr size_t AN_COORD  = 0;
constexpr size_t AN_FEAT   = AN_COORD  + (size_t)B_*6*N_;
constexpr size_t AK_COORD  = AN_FEAT   + (size_t)B_*128*N_;
constexpr size_t AK_FEAT   = AK_COORD  + (size_t)B_*6*K_;
constexpr size_t SN_STATS  = AK_FEAT   + (size_t)B_*128*K_;     // [b][c][2] sum,sumsq over n
constexpr size_t SK_STATS  = SN_STATS  + (size_t)B_*F_*2;       // [b][c][2] sum,sumsq over k
constexpr size_t SCALE1    = SK_STATS  + (size_t)B_*F_*2;       // [134]
constexpr size_t SHIFT1    = SCALE1    + F_;                    // [134]
constexpr size_t GMAX      = SHIFT1    + F_;                    // [b][o][n]
constexpr size_t GMIN      = GMAX      + (size_t)B_*D_*N_;
constexpr size_t FPART_SUM = GMIN      + (size_t)B_*D_*N_;      // [512 blocks][64]
constexpr size_t FPART_SQ  = FPART_SUM + 512*64;
constexpr size_t FSTATS    = FPART_SQ  + 512*64;                // [o][2]
constexpr size_t AGG       = FSTATS    + 128;                   // [b][o][n]
constexpr size_t OUTPRE    = AGG       + (size_t)B_*D_*N_;      // [b][o][n]
constexpr size_t OSTATS    = OUTPRE    + (size_t)B_*D_*N_;      // [o][2]

__device__ __forceinline__ float ldAn(const float* ws, int b, int c, int n) {
  return (c < 6) ? ws[AN_COORD + ((size_t)(b*6 + c))*N_ + n]
                 : ws[AN_FEAT  + ((size_t)(b*128 + (c-6)))*N_ + n];
}
__device__ __forceinline__ float ldAk(const float* ws, int b, int c, int k) {
  return (c < 6) ? ws[AK_COORD + (size_t)(b*6 + c)*K_ + k]
                 : ws[AK_FEAT  + (size_t)(b*128 + (c-6))*K_ + k];
}

__device__ __forceinline__ v8f wmma4(v2f a, v2f b, v8f c) {
  // D = A(16x4,f32) * B(4x16,f32) + C(16x16,f32)
  return __builtin_amdgcn_wmma_f32_16x16x4_f32(false, a, false, b, (short)0, c, false, false);
}

// ---- Stage 1: per-n projections An[b,c,n] (left halves of W_coord/W_feat) ----
__global__ __launch_bounds__(256) void k_stage1_n(const float* __restrict__ coords,
                                                  const float* __restrict__ features,
                                                  const float* __restrict__ Wc,
                                                  const float* __restrict__ Wf,
                                                  float* __restrict__ ws) {
  int n = blockIdx.x * 256 + threadIdx.x;
  int o = blockIdx.y;      // 0..133
  int b = blockIdx.z;
  float acc = 0.f;
  if (o < 6) {
    #pragma unroll
    for (int c = 0; c < 3; ++c) acc += Wc[o*6 + c] * coords[((size_t)(b*3 + c))*N_ + n];
    ws[AN_COORD + ((size_t)(b*6 + o))*N_ + n] = acc;
  } else {
    int oo = o - 6;
    #pragma unroll 8
    for (int d = 0; d < 64; ++d) acc += Wf[oo*128 + d] * features[((size_t)(b*64 + d))*N_ + n];
    ws[AN_FEAT + ((size_t)(b*128 + oo))*N_ + n] = acc;
  }
}

// ---- Stage 1: gathered per-k projections Ak[b,c,k] (right halves) ----
__global__ __launch_bounds__(256) void k_stage1_k(const float* __restrict__ coords,
                                                  const float* __restrict__ features,
                                                  const int* __restrict__ ridx,
                                                  const float* __restrict__ Wc,
                                                  const float* __restrict__ Wf,
                                                  float* __restrict__ ws) {
  int flat = blockIdx.x * 256 + threadIdx.x;
  if (flat >= B_ * F_ * K_) return;
  int b = flat / (F_ * K_);
  int rem = flat % (F_ * K_);
  int o = rem / K_, k = rem % K_;
  int id = ridx[b*K_ + k];
  float acc = 0.f;
  if (o < 6) {
    #pragma unroll
    for (int c = 0; c < 3; ++c) acc += Wc[o*6 + 3 + c] * coords[((size_t)(b*3 + c))*N_ + id];
    ws[AK_COORD + (size_t)(b*6 + o)*K_ + k] = acc;
  } else {
    int oo = o - 6;
    for (int d = 0; d < 64; ++d) acc += Wf[oo*128 + 64 + d] * features[((size_t)(b*64 + d))*N_ + id];
    ws[AK_FEAT + (size_t)(b*128 + oo)*K_ + k] = acc;
  }
}

// ---- Stage 1: per-(b,c) sums for separable BN statistics ----
__global__ __launch_bounds__(256) void k_stage1_stats(float* __restrict__ ws) {
  __shared__ float red[512];
  int bc = blockIdx.x;       // b*134 + c
  int b = bc / F_, c = bc % F_;
  int tid = threadIdx.x;
  float s = 0.f, s2 = 0.f;
  for (int i = tid; i < N_; i += 256) { float v = ldAn(ws, b, c, i); s += v; s2 += v*v; }
  red[tid] = s; red[256 + tid] = s2;
  __syncthreads();
  for (int st = 128; st > 0; st >>= 1) {
    if (tid < st) { red[tid] += red[tid + st]; red[256 + tid] += red[256 + tid + st]; }
    __syncthreads();
  }
  if (tid == 0) { ws[SN_STATS + bc*2] = red[0]; ws[SN_STATS + bc*2 + 1] = red[256]; }
  __syncthreads();
  float a = 0.f, a2 = 0.f;
  if (tid < K_) { float v = ldAk(ws, b, c, tid); a = v; a2 = v*v; }
  red[tid] = a; red[256 + tid] = a2;
  __syncthreads();
  for (int st = 128; st > 0; st >>= 1) {
    if (tid < st) { red[tid] += red[tid + st]; red[256 + tid] += red[256 + tid + st]; }
    __syncthreads();
  }
  if (tid == 0) { ws[SK_STATS + bc*2] = red[0]; ws[SK_STATS + bc*2 + 1] = red[256]; }
}

// ---- Stage 1: fold BN(mean,var,gamma,beta) -> scale1/shift1 per channel ----
__global__ void k_stage1_scale(const float* __restrict__ gc, const float* __restrict__ bec,
                               const float* __restrict__ gf, const float* __restrict__ bef,
                               float* __restrict__ ws) {
  int c = threadIdx.x;
  if (c >= F_) return;
  float sum = 0.f, sq = 0.f;
  for (int b = 0; b < B_; ++b) {
    float sn  = ws[SN_STATS + (size_t)(b*F_ + c)*2];
    float sn2 = ws[SN_STATS + (size_t)(b*F_ + c)*2 + 1];
    float sk  = ws[SK_STATS + (size_t)(b*F_ + c)*2];
    float sk2 = ws[SK_STATS + (size_t)(b*F_ + c)*2 + 1];
    // sum over (k,n) of (An+Ak) and (An+Ak)^2
    sum += (float)K_ * sn + (float)N_ * sk;
    sq  += (float)K_ * sn2 + 2.f * sn * sk + (float)N_ * sk2;
  }
  float cnt = (float)B_ * K_ * N_;
  float mean = sum / cnt;
  float var  = sq / cnt - mean * mean;
  float g  = (c < 6) ? gc[c]  : gf[c - 6];
  float be = (c < 6) ? bec[c] : bef[c - 6];
  float sc = g * rsqrtf(var + EPS_);
  ws[SCALE1 + c] = sc;
  ws[SHIFT1 + c] = be - mean * sc;
}

// ---- Stage 2: fused 64x134 GEMM over on-the-fly combined tensor (WMMA f32) ----
// Grid: 512 blocks = (b, 16-n tile). Block 128 thr = 4 waves; wave w -> rows [16w,16w+16), all 64 k.
// Two independent accumulator chains (even/odd s) to decouple consecutive WMMAs.
__global__ __launch_bounds__(128) void k_fused(const float* __restrict__ Wfuse,
                                               float* __restrict__ ws) {
  __shared__ float Skl[136 * 64];   // scale1[c]*Ak + shift1[c], padded channels = 0
  int blk = blockIdx.x;
  int b = blk >> 8;
  int n0 = (blk & 255) << 4;
  int tid = threadIdx.x;

  for (int i = tid; i < 136 * 64; i += 128) {
    int c = i >> 6, k = i & 63;
    float v = 0.f;
    if (c < F_) v = ws[SCALE1 + c] * ldAk(ws, b, c, k) + ws[SHIFT1 + c];
    Skl[i] = v;
  }
  __syncthreads();

  int w = tid >> 5, lane = tid & 31, half = lane >> 4, j = lane & 15;
  int n = n0 + j;
  int orow = w * 16 + j;             // A-matrix row = lane%16

  // Hoist k-invariant operands into registers.
  v2f aW[S_], bn[S_];
  #pragma unroll
  for (int s = 0; s < S_; ++s) {
    int c0 = 4 * s + 2 * half;       // A/B lane layout: VGPR0 -> K0(lo)/K2(hi), VGPR1 -> K1/K3
    v2f a; a.x = 0.f; a.y = 0.f;
    v2f bb; bb.x = 0.f; bb.y = 0.f;
    if (c0 < F_)     { a.x = Wfuse[orow * F_ + c0];     bb.x = ws[SCALE1 + c0]     * ldAn(ws, b, c0, n); }
    if (c0 + 1 < F_) { a.y = Wfuse[orow * F_ + c0 + 1]; bb.y = ws[SCALE1 + c0 + 1] * ldAn(ws, b, c0 + 1, n); }
    aW[s] = a; bn[s] = bb;
  }

  v8f vmax, vmin, vsum, vsq;
  #pragma unroll
  for (int v = 0; v < 8; ++v) { vmax[v] = -INFINITY; vmin[v] = INFINITY; vsum[v] = 0.f; vsq[v] = 0.f; }

  for (int k = 0; k < K_; ++k) {
    v8f acc0, acc1;
    #pragma unroll
    for (int v = 0; v < 8; ++v) { acc0[v] = 0.f; acc1[v] = 0.f; }
    #pragma unroll
    for (int s = 0; s < S_; ++s) {
      int c0 = 4 * s + 2 * half;
      v2f bf;
      bf.x = fmaxf(bn[s].x + Skl[c0 * 64 + k], 0.f);       // relu(BN(An+Ak))
      bf.y = fmaxf(bn[s].y + Skl[(c0 + 1) * 64 + k], 0.f);
      if (s & 1) acc1 = wmma4(aW[s], bf, acc1);
      else       acc0 = wmma4(aW[s], bf, acc0);
    }
    #pragma unroll
    for (int v = 0; v < 8; ++v) {
      float g = acc0[v] + acc1[v];
      vmax[v] = fmaxf(vmax[v], g);
      vmin[v] = fminf(vmin[v], g);
      vsum[v] += g;
      vsq[v]  += g * g;
    }
  }

  // C/D layout: VGPR v -> row v (lanes 0-15) / v+8 (lanes 16-31); col = lane%16.
  #pragma unroll
  for (int v = 0; v < 8; ++v) {
    int o = w * 16 + v + half * 8;
    size_t idx = ((size_t)(b * 64 + o)) * N_ + n;
    ws[GMAX + idx] = vmax[v];
    ws[GMIN + idx] = vmin[v];
    float s1 = vsum[v], s2 = vsq[v];
    #pragma unroll
    for (int m = 1; m <= 8; m <<= 1) { s1 += __shfl_xor(s1, m, 32); s2 += __shfl_xor(s2, m, 32); }
    if (j == 0) {
      ws[FPART_SUM + (size_t)blk * 64 + o] = s1;
      ws[FPART_SQ  + (size_t)blk * 64 + o] = s2;
    }
  }
}

// ---- Deterministic reduction of fused BN partials ----
__global__ void k_fstats(float* __restrict__ ws) {
  int o = threadIdx.x;
  if (o >= 64) return;
  float s = 0.f, s2 = 0.f;
  for (int blk = 0; blk < 512; ++blk) {
    s  += ws[FPART_SUM + (size_t)blk * 64 + o];
    s2 += ws[FPART_SQ  + (size_t)blk * 64 + o];
  }
  ws[FSTATS + o*2] = s;
  ws[FSTATS + o*2 + 1] = s2;
}

// ---- aggregated = max_k relu(BN(G)) via sign-aware Gmax/Gmin ----
__global__ __launch_bounds__(256) void k_agg(const float* __restrict__ gfuse,
                                             const float* __restrict__ bfuse,
                                             float* __restrict__ ws) {
  int gid = blockIdx.x * 256 + threadIdx.x;   // b*64*N + o*N + n
  if (gid >= B_ * D_ * N_) return;
  int o = (gid >> 12) & 63;
  float cnt = (float)B_ * K_ * N_;
  float mean = ws[FSTATS + o*2] / cnt;
  float var  = ws[FSTATS + o*2 + 1] / cnt - mean * mean;
  float sc = gfuse[o] * rsqrtf(var + EPS_);
  float sh = bfuse[o] - mean * sc;
  float g = (sc >= 0.f) ? ws[GMAX + gid] : ws[GMIN + gid];
  ws[AGG + gid] = fmaxf(sc * g + sh, 0.f);
}

// ---- W_out (64x64) @ aggregated, WMMA f32, two accumulator chains ----
__global__ __launch_bounds__(128) void k_out_mm(const float* __restrict__ Wout,
                                                float* __restrict__ ws) {
  int blk = blockIdx.x;              // 512 = (b, 16-n tile)
  int b = blk >> 8;
  int n0 = (blk & 255) << 4;
  int tid = threadIdx.x;
  int w = tid >> 5, lane = tid & 31, half = lane >> 4, j = lane & 15;
  int n = n0 + j;
  int orow = w * 16 + j;
  v2f aW[16], bf[16];
  #pragma unroll
  for (int s = 0; s < 16; ++s) {
    int c0 = 4 * s + 2 * half;
    aW[s].x = Wout[orow * 64 + c0];
    aW[s].y = Wout[orow * 64 + c0 + 1];
    bf[s].x = ws[AGG + ((size_t)(b * 64 + c0)) * N_ + n];
    bf[s].y = ws[AGG + ((size_t)(b * 64 + c0 + 1)) * N_ + n];
  }
  v8f acc0, acc1;
  #pragma unroll
  for (int v = 0; v < 8; ++v) { acc0[v] = 0.f; acc1[v] = 0.f; }
  #pragma unroll
  for (int s = 0; s < 16; s += 2) {
    acc0 = wmma4(aW[s],     bf[s],     acc0);
    acc1 = wmma4(aW[s + 1], bf[s + 1], acc1);
  }
  #pragma unroll
  for (int v = 0; v < 8; ++v) {
    int o = w * 16 + v + half * 8;
    ws[OUTPRE + ((size_t)(b * 64 + o)) * N_ + n] = acc0[v] + acc1[v];
  }
}

// ---- out-BN statistics (per o over b,n), deterministic tree ----
__global__ __launch_bounds__(256) void k_ostats(float* __restrict__ ws) {
  __shared__ float red[512];
  int o = blockIdx.x, tid = threadIdx.x;
  float s = 0.f, s2 = 0.f;
  for (int i = tid; i < B_ * N_; i += 256) {
    int b = i >> 12, n = i & 4095;
    float v = ws[OUTPRE + ((size_t)(b * 64 + o)) * N_ + n];
    s += v; s2 += v * v;
  }
  red[tid] = s; red[256 + tid] = s2;
  __syncthreads();
  for (int st = 128; st > 0; st >>= 1) {
    if (tid < st) { red[tid] += red[tid + st]; red[256 + tid] += red[256 + tid + st]; }
    __syncthreads();
  }
  if (tid == 0) { ws[OSTATS + o*2] = red[0]; ws[OSTATS + o*2 + 1] = red[256]; }
}

// ---- final: [features + relu(BN(out_pre)) ; coords] ----
__global__ __launch_bounds__(256) void k_final(const float* __restrict__ coords,
                                               const float* __restrict__ features,
                                               const float* __restrict__ gout,
                                               const float* __restrict__ bout,
                                               const float* __restrict__ ws,
                                               float* __restrict__ out) {
  int gid = blockIdx.x * 256 + threadIdx.x;   // B*67*N
  if (gid >= B_ * 67 * N_) return;
  int n = gid & 4095;
  int r = (gid >> 12) % 67;
  int b = gid / (67 * N_);
  float val;
  if (r < 64) {
    float cnt = (float)B_ * N_;
    float mean = ws[OSTATS + r*2] / cnt;
    float var  = ws[OSTATS + r*2 + 1] / cnt - mean * mean;
    float sc = gout[r] * rsqrtf(var + EPS_);
    float sh = bout[r] - mean * sc;
    float op = ws[OUTPRE + ((size_t)(b * 64 + r)) * N_ + n];
    val = features[((size_t)(b * 64 + r)) * N_ + n] + fmaxf(sc * op + sh, 0.f);
  } else {
    val = coords[((size_t)(b * 3 + (r - 64))) * N_ + n];
  }
  out[gid] = val;
}

extern "C" void kernel_launch(void* const* d_in, const int* in_sizes, int n_in,
                              void* d_out, int out_size, void* d_ws, size_t ws_size,
                              hipStream_t stream) {
  const float* coords   = (const float*)d_in[0];
  const float* features = (const float*)d_in[1];
  const int*   ridx     = (const int*)d_in[2];
  const float* Wc       = (const float*)d_in[3];
  const float* gc       = (const float*)d_in[4];
  const float* bec      = (const float*)d_in[5];
  const float* Wf       = (const float*)d_in[6];
  const float* gf       = (const float*)d_in[7];
  const float* bef      = (const float*)d_in[8];
  const float* Wfuse    = (const float*)d_in[9];
  const float* gfuse    = (const float*)d_in[10];
  const float* bfuse    = (const float*)d_in[11];
  const float* Wout     = (const float*)d_in[12];
  const float* gout     = (const float*)d_in[13];
  const float* bout     = (const float*)d_in[14];
  float* out = (float*)d_out;
  float* ws  = (float*)d_ws;

  k_stage1_n<<<dim3(N_/256, F_, B_), 256, 0, stream>>>(coords, features, Wc, Wf, ws);
  k_stage1_k<<<(B_*F_*K_ + 255)/256, 256, 0, stream>>>(coords, features, ridx, Wc, Wf, ws);
  k_stage1_stats<<<B_*F_, 256, 0, stream>>>(ws);
  k_stage1_scale<<<1, 160, 0, stream>>>(gc, bec, gf, bef, ws);
  k_fused<<<B_*(N_/16), 128, 0, stream>>>(Wfuse, ws);
  k_fstats<<<1, 64, 0, stream>>>(ws);
  k_agg<<<(B_*D_*N_)/256, 256, 0, stream>>>(gfuse, bfuse, ws);
  k_out_mm<<<B_*(N_/16), 128, 0, stream>>>(Wout, ws);
  k_ostats<<<64, 256, 0, stream>>>(ws);
  k_final<<<(B_*67*N_)/256, 256, 0, stream>>>(coords, features, gout, bout, ws, out);
}